// Simplify_S6_ScanFeat_62079457296616
// MI455X (gfx1250) — compile-verified
//
#include <hip/hip_runtime.h>
#include <hip/hip_bf16.h>
#include <math.h>

// Problem constants (from reference)
#define BATCH   4
#define D_IN    2048
#define SEQ     4096
#define NSTATE  16
#define RANK    128
#define E_TOT   (RANK + 2 * NSTATE)   // 160

typedef __attribute__((ext_vector_type(2))) float v2f;
typedef __attribute__((ext_vector_type(8))) float v8f;

// ---------------------------------------------------------------------------
// GEMM 1: x_dbl[b,e,l] = sum_d x_proj_w[e,d] * x[b,d,l]
// One wave computes a 16(M=e) x 64(N=l) tile using V_WMMA_F32_16X16X4_F32.
// A (x_proj_w) is row-major E x D; B is x[b] row-major D x L.
// A-frag layout (16x4 f32): lanes 0-15 hold K={0,1}, lanes 16-31 hold K={2,3}
//   -> per-lane float2 load at row (e0+lane&15), col k+2*(lane>>4).
// B-frag layout (4x16 f32): lanes 0-15 col N=0..15 K={0,1}; lanes 16-31 K={2,3}.
// C/D layout: VGPR j -> row m = j + 8*(lane>>4), col n = lane&15.
// ---------------------------------------------------------------------------
__global__ __launch_bounds__(32) void s6_gemm_xdbl(
    const float* __restrict__ x,       // [B, D_IN, SEQ]
    const float* __restrict__ w,       // [E_TOT, D_IN]
    float* __restrict__ xdbl)          // [B, E_TOT, SEQ]
{
    const int lane = threadIdx.x;
    const int lo   = lane & 15;
    const int hi   = lane >> 4;              // 0 or 1
    const int e0   = blockIdx.x * 16;
    const int n0   = blockIdx.y * 64;
    const int b    = blockIdx.z;

    const float* __restrict__ xb   = x + (size_t)b * D_IN * SEQ;
    const float* __restrict__ arow = w + (size_t)(e0 + lo) * D_IN + 2 * hi;

    v8f acc[4] = {v8f{}, v8f{}, v8f{}, v8f{}};

    for (int k = 0; k < D_IN; k += 4) {
        v2f a = *(const v2f*)(arow + k);
        const float* __restrict__ br0 = xb + (size_t)(k + 2 * hi) * SEQ + n0 + lo;
        const float* __restrict__ br1 = br0 + SEQ;
#pragma unroll
        for (int q = 0; q < 4; ++q) {
            v2f bf;
            bf.x = br0[16 * q];
            bf.y = br1[16 * q];
            acc[q] = __builtin_amdgcn_wmma_f32_16x16x4_f32(
                false, a, false, bf, (short)0, acc[q], false, false);
        }
    }

    float* __restrict__ orow = xdbl + ((size_t)b * E_TOT + e0) * SEQ + n0;
#pragma unroll
    for (int j = 0; j < 8; ++j) {
        const size_t m = (size_t)(j + 8 * hi) * SEQ;
#pragma unroll
        for (int q = 0; q < 4; ++q)
            orow[m + 16 * q + lo] = acc[q][j];
    }
}

// ---------------------------------------------------------------------------
// GEMM 2 + bias + softplus:
// delta[b,d,l] = softplus( sum_r dt_proj_w[d,r] * x_dbl[b,r,l] + dt_proj_b[d] )
// ---------------------------------------------------------------------------
__global__ __launch_bounds__(32) void s6_gemm_delta(
    const float* __restrict__ xdbl,    // [B, E_TOT, SEQ] (rows 0..RANK-1 used)
    const float* __restrict__ w,       // [D_IN, RANK]
    const float* __restrict__ bias,    // [D_IN]
    float* __restrict__ delta)         // [B, D_IN, SEQ]
{
    const int lane = threadIdx.x;
    const int lo   = lane & 15;
    const int hi   = lane >> 4;
    const int d0   = blockIdx.x * 16;
    const int n0   = blockIdx.y * 64;
    const int b    = blockIdx.z;

    const float* __restrict__ xb   = xdbl + (size_t)b * E_TOT * SEQ;
    const float* __restrict__ arow = w + (size_t)(d0 + lo) * RANK + 2 * hi;

    v8f acc[4] = {v8f{}, v8f{}, v8f{}, v8f{}};

    for (int k = 0; k < RANK; k += 4) {
        v2f a = *(const v2f*)(arow + k);
        const float* __restrict__ br0 = xb + (size_t)(k + 2 * hi) * SEQ + n0 + lo;
        const float* __restrict__ br1 = br0 + SEQ;
#pragma unroll
        for (int q = 0; q < 4; ++q) {
            v2f bf;
            bf.x = br0[16 * q];
            bf.y = br1[16 * q];
            acc[q] = __builtin_amdgcn_wmma_f32_16x16x4_f32(
                false, a, false, bf, (short)0, acc[q], false, false);
        }
    }

    float* __restrict__ orow = delta + ((size_t)b * D_IN + d0) * SEQ + n0;
#pragma unroll
    for (int j = 0; j < 8; ++j) {
        const int   m  = j + 8 * hi;
        const float bz = bias[d0 + m];
#pragma unroll
        for (int q = 0; q < 4; ++q) {
            float z  = acc[q][j] + bz;
            float sp = (z > 20.0f) ? z : log1pf(__expf(z));
            orow[(size_t)m * SEQ + 16 * q + lo] = sp;
        }
    }
}

// ---------------------------------------------------------------------------
// Selective scan. One thread per (b, d) channel; 16 states in registers.
// B_t / C_t tiles (shared by all channels of a batch) staged in LDS.
// ---------------------------------------------------------------------------
#define SCAN_BD 128   // threads per block == channels per block == timestep tile
__global__ __launch_bounds__(SCAN_BD) void s6_scan(
    const float* __restrict__ x,       // [B, D_IN, SEQ]
    const float* __restrict__ A_log,   // [D_IN, NSTATE]
    const float* __restrict__ Dw,      // [D_IN]
    const float* __restrict__ xdbl,    // [B, E_TOT, SEQ]  (rows RANK.. for B/C)
    const float* __restrict__ delta,   // [B, D_IN, SEQ]
    float* __restrict__ out)           // [B, D_IN, SEQ]
{
    __shared__ float sB[NSTATE][SCAN_BD];
    __shared__ float sC[NSTATE][SCAN_BD];

    const int tid = threadIdx.x;
    const int d   = blockIdx.x * SCAN_BD + tid;
    const int b   = blockIdx.y;

    float Arow[NSTATE];
#pragma unroll
    for (int n = 0; n < NSTATE; ++n)
        Arow[n] = -__expf(A_log[(size_t)d * NSTATE + n]);
    const float Dd = Dw[d];

    float h[NSTATE];
#pragma unroll
    for (int n = 0; n < NSTATE; ++n) h[n] = 0.0f;

    const size_t ch = (size_t)b * D_IN * SEQ + (size_t)d * SEQ;
    const float* __restrict__ drow = delta + ch;
    const float* __restrict__ urow = x + ch;
    float*       __restrict__ yrow = out + ch;
    const float* __restrict__ Bsrc = xdbl + ((size_t)b * E_TOT + RANK) * SEQ;
    const float* __restrict__ Csrc = xdbl + ((size_t)b * E_TOT + RANK + NSTATE) * SEQ;

    for (int l0 = 0; l0 < SEQ; l0 += SCAN_BD) {
        __syncthreads();   // protect LDS reuse from previous tile
#pragma unroll
        for (int k = 0; k < NSTATE; ++k) {
            sB[k][tid] = Bsrc[(size_t)k * SEQ + l0 + tid];   // coalesced along l
            sC[k][tid] = Csrc[(size_t)k * SEQ + l0 + tid];
        }
        __syncthreads();

        for (int t = 0; t < SCAN_BD; t += 4) {
            const float4 dv = *(const float4*)(drow + l0 + t);
            const float4 uv = *(const float4*)(urow + l0 + t);
            float dts[4] = {dv.x, dv.y, dv.z, dv.w};
            float us[4]  = {uv.x, uv.y, uv.z, uv.w};
            float ys[4];
#pragma unroll
            for (int s = 0; s < 4; ++s) {
                const float dt = dts[s];
                const float du = dt * us[s];
                float y = 0.0f;
#pragma unroll
                for (int n = 0; n < NSTATE; ++n) {
                    const float dA = __expf(dt * Arow[n]);
                    h[n] = h[n] * dA + du * sB[n][t + s];
                    y += h[n] * sC[n][t + s];
                }
                ys[s] = y + us[s] * Dd;
            }
            float4 yv = {ys[0], ys[1], ys[2], ys[3]};
            *(float4*)(yrow + l0 + t) = yv;
        }
    }
}

// ---------------------------------------------------------------------------
// Launch
// ---------------------------------------------------------------------------
extern "C" void kernel_launch(void* const* d_in, const int* in_sizes, int n_in,
                              void* d_out, int out_size, void* d_ws, size_t ws_size,
                              hipStream_t stream) {
    const float* x    = (const float*)d_in[0];   // [B, D_IN, SEQ]
    const float* Alog = (const float*)d_in[1];   // [D_IN, NSTATE]
    const float* Dw   = (const float*)d_in[2];   // [D_IN]
    const float* xpw  = (const float*)d_in[3];   // [E_TOT, D_IN]
    const float* dtw  = (const float*)d_in[4];   // [D_IN, RANK]
    const float* dtb  = (const float*)d_in[5];   // [D_IN]
    float* out = (float*)d_out;

    float* xdbl  = (float*)d_ws;                             // B*E_TOT*SEQ floats
    float* delta = xdbl + (size_t)BATCH * E_TOT * SEQ;       // B*D_IN*SEQ floats

    dim3 g1(E_TOT / 16, SEQ / 64, BATCH);
    s6_gemm_xdbl<<<g1, 32, 0, stream>>>(x, xpw, xdbl);

    dim3 g2(D_IN / 16, SEQ / 64, BATCH);
    s6_gemm_delta<<<g2, 32, 0, stream>>>(xdbl, dtw, dtb, delta);

    dim3 g3(D_IN / SCAN_BD, BATCH);
    s6_scan<<<g3, SCAN_BD, 0, stream>>>(x, Alog, Dw, xdbl, delta, out);
}